// GraphConv_wrapper_75900662055243
// MI455X (gfx1250) — compile-verified
//
#include <hip/hip_runtime.h>

constexpr int N_NODES = 50000;   // divisible by 16 -> no M predication
constexpr int N_EDGES = 800000;
constexpr int D_IN    = 100;
constexpr int D_OUT   = 200;
constexpr int NPAD    = 208;     // 13 * 16 padded N
constexpr int NT      = 13;      // N tiles per row stripe
constexpr int TPB     = 256;

typedef __attribute__((ext_vector_type(2))) float v2f;
typedef __attribute__((ext_vector_type(8))) float v8f;

// ---------------------------------------------------------------------------
// Zero a float buffer (grid-stride).
// ---------------------------------------------------------------------------
__global__ void zero_f32(float* __restrict__ p, int n) {
  int i = blockIdx.x * blockDim.x + threadIdx.x;
  int stride = gridDim.x * blockDim.x;
  for (; i < n; i += stride) p[i] = 0.0f;
}

// ---------------------------------------------------------------------------
// Pad [K x 200] row-major weight into [K x 208] (zeros in cols 200..207).
// ---------------------------------------------------------------------------
template <int K>
__global__ void pad_weight(const float* __restrict__ w, float* __restrict__ wp) {
  int i = blockIdx.x * blockDim.x + threadIdx.x;
  if (i >= K * NPAD) return;
  int r = i / NPAD;
  int c = i - r * NPAD;
  wp[i] = (c < D_OUT) ? w[r * D_OUT + c] : 0.0f;
}

__global__ void pad_bias(const float* __restrict__ b, float* __restrict__ bp) {
  int i = threadIdx.x;
  if (i < NPAD) bp[i] = (i < D_OUT) ? b[i] : 0.0f;
}

// ---------------------------------------------------------------------------
// Edge aggregation: agg[dst[e]][c..c+3] += x[src[e]][c..c+3]
// One thread per (edge, 4-channel chunk). 128-bit gather, 4 f32 atomics.
// ---------------------------------------------------------------------------
template <int D>
__global__ void aggregate(const float* __restrict__ x,
                          const int* __restrict__ src,
                          const int* __restrict__ dst,
                          float* __restrict__ agg) {
  constexpr int CH = D / 4;                       // const divisor -> mul/shift
  long long tid = (long long)blockIdx.x * blockDim.x + threadIdx.x;
  if (tid >= (long long)N_EDGES * CH) return;
  int e = (int)(tid / CH);
  int c = (int)(tid - (long long)e * CH) << 2;
  int s = src[e];
  int t = dst[e];
  const float4 v = *(const float4*)(x + (long long)s * D + c);
  float* o = agg + (long long)t * D + c;
  atomicAdd(o + 0, v.x);
  atomicAdd(o + 1, v.y);
  atomicAdd(o + 2, v.z);
  atomicAdd(o + 3, v.w);
}

// ---------------------------------------------------------------------------
// out[50000 x 200] = agg @ w_rel + bias + h @ w_root   (f32, row-major)
// One wave owns a 16 x 208 output stripe: 13 v8f accumulators (104 VGPRs),
// A loaded once per k-step (global_load_b64) and reused across 13
// V_WMMA_F32_16X16X4_F32. Weights/bias pre-padded -> unconditional loads.
// __launch_bounds__(TPB, 1) lifts the VGPR cap so accumulators do NOT spill.
// ---------------------------------------------------------------------------
template <int K>
__global__ void __launch_bounds__(TPB, 1)
gemm_dual_wmma(const float* __restrict__ agg,
               const float* __restrict__ h,
               const float* __restrict__ wrelp,
               const float* __restrict__ wrootp,
               const float* __restrict__ biasp,
               float* __restrict__ out) {
  const int lane = threadIdx.x & 31;
  const int wave = threadIdx.x >> 5;
  const int m_tile = blockIdx.x * (TPB / 32) + wave;
  if (m_tile * 16 >= N_NODES) return;   // wave-uniform: EXEC stays all-1s

  const int m_base = m_tile << 4;
  const int l16    = lane & 15;
  const int khi    = (lane >> 4) << 1;  // 0 for lanes 0-15, 2 for lanes 16-31
  const int mrow   = m_base + l16;      // always < N_NODES (50000 % 16 == 0)

  v8f acc[NT];
#pragma unroll
  for (int t = 0; t < NT; t++) acc[t] = (v8f){0, 0, 0, 0, 0, 0, 0, 0};

  // ---- agg @ w_rel ----
  {
    const float* arow = agg + (long long)mrow * K + khi;
    for (int k = 0; k < K; k += 4) {
      const v2f a = *(const v2f*)(arow + k);
      const float* brow = wrelp + (k + khi) * NPAD + l16;
#pragma unroll
      for (int t = 0; t < NT; t++) {
        v2f b;
        b.x = brow[t * 16];
        b.y = brow[NPAD + t * 16];
        acc[t] = __builtin_amdgcn_wmma_f32_16x16x4_f32(false, a, false, b,
                                                       (short)0, acc[t],
                                                       false, false);
      }
    }
  }
  // ---- h @ w_root ----
  {
    const float* arow = h + (long long)mrow * K + khi;
    for (int k = 0; k < K; k += 4) {
      const v2f a = *(const v2f*)(arow + k);
      const float* brow = wrootp + (k + khi) * NPAD + l16;
#pragma unroll
      for (int t = 0; t < NT; t++) {
        v2f b;
        b.x = brow[t * 16];
        b.y = brow[NPAD + t * 16];
        acc[t] = __builtin_amdgcn_wmma_f32_16x16x4_f32(false, a, false, b,
                                                       (short)0, acc[t],
                                                       false, false);
      }
    }
  }

  // ---- epilogue: bias + predicated store (only tile 12 is partial) ----
  const int mofs = (lane >> 4) << 3;    // +8 rows for lanes 16-31
#pragma unroll
  for (int t = 0; t < NT; t++) {
    const int ncol = t * 16 + l16;
    if (ncol < D_OUT) {
      const float bv = biasp[ncol];
#pragma unroll
      for (int r = 0; r < 8; r++) {
        out[(long long)(m_base + r + mofs) * D_OUT + ncol] = acc[t][r] + bv;
      }
    }
  }
}

// ---------------------------------------------------------------------------
// mean_h[c] = 0.1 * sum_{i<10} h[i][c]
// ---------------------------------------------------------------------------
__global__ void mean_front10(const float* __restrict__ h,
                             float* __restrict__ mean_h) {
  int c = blockIdx.x * blockDim.x + threadIdx.x;
  if (c >= D_OUT) return;
  float s = 0.0f;
  for (int i = 0; i < 10; i++) s += h[i * D_OUT + c];
  mean_h[c] = s * 0.1f;
}

// ---------------------------------------------------------------------------
// sum_agg[c] += 0.1 * h[src[e]][c]  for every edge with dst[e] < 10
// ---------------------------------------------------------------------------
__global__ void agg_front10(const float* __restrict__ h,
                            const int* __restrict__ src,
                            const int* __restrict__ dst,
                            float* __restrict__ sum_agg) {
  int e = blockIdx.x * blockDim.x + threadIdx.x;
  if (e >= N_EDGES) return;
  if (dst[e] >= 10) return;
  const float* row = h + (long long)src[e] * D_OUT;
  for (int c = 0; c < D_OUT; c++) atomicAdd(&sum_agg[c], row[c] * 0.1f);
}

// ---------------------------------------------------------------------------
// out[n] = sum_k mean_agg[k]*w_rel[k][n] + bias[n] + sum_k mean_h[k]*w_root[k][n]
// ---------------------------------------------------------------------------
__global__ void final_gemv(const float* __restrict__ mean_agg,
                           const float* __restrict__ mean_h,
                           const float* __restrict__ w_rel,
                           const float* __restrict__ w_root,
                           const float* __restrict__ bias,
                           float* __restrict__ out) {
  int n = blockIdx.x * blockDim.x + threadIdx.x;
  if (n >= D_OUT) return;
  float acc = bias[n];
  for (int k = 0; k < D_OUT; k++) {
    acc = fmaf(mean_agg[k], w_rel[k * D_OUT + n], acc);
    acc = fmaf(mean_h[k],   w_root[k * D_OUT + n], acc);
  }
  out[n] = acc;
}

// ---------------------------------------------------------------------------
extern "C" void kernel_launch(void* const* d_in, const int* in_sizes, int n_in,
                              void* d_out, int out_size, void* d_ws, size_t ws_size,
                              hipStream_t stream) {
  const float* x       = (const float*)d_in[0];
  const int*   edges   = (const int*)d_in[1];     // [2, N_EDGES]
  const float* w1_rel  = (const float*)d_in[2];
  const float* b1_rel  = (const float*)d_in[3];
  const float* w1_root = (const float*)d_in[4];
  const float* w2_rel  = (const float*)d_in[5];
  const float* b2_rel  = (const float*)d_in[6];
  const float* w2_root = (const float*)d_in[7];
  const float* w3_rel  = (const float*)d_in[8];
  const float* b3_rel  = (const float*)d_in[9];
  const float* w3_root = (const float*)d_in[10];
  float* out = (float*)d_out;

  const int* src = edges;
  const int* dst = edges + N_EDGES;

  // ---- workspace carve-up (floats) ----
  float* ws = (float*)d_ws;
  float* h1      = ws;                                 // 10,000,000
  float* h2      = h1 + (long long)N_NODES * D_OUT;    // 10,000,000
  float* agg     = h2 + (long long)N_NODES * D_OUT;    // 10,000,000
  float* w1relp  = agg + (long long)N_NODES * D_OUT;   // 100*208
  float* w1rootp = w1relp  + D_IN * NPAD;
  float* w2relp  = w1rootp + D_IN * NPAD;              // 200*208
  float* w2rootp = w2relp  + D_OUT * NPAD;
  float* b1p     = w2rootp + D_OUT * NPAD;             // 208
  float* b2p     = b1p + NPAD;                         // 208
  float* mean_h  = b2p + NPAD;                         // 200
  float* sum_agg = mean_h + D_OUT;                     // 200

  // ---- pad weights/bias to width 208 (removes all B-load predication) ----
  pad_weight<D_IN> <<<(D_IN * NPAD + TPB - 1) / TPB, TPB, 0, stream>>>(w1_rel,  w1relp);
  pad_weight<D_IN> <<<(D_IN * NPAD + TPB - 1) / TPB, TPB, 0, stream>>>(w1_root, w1rootp);
  pad_weight<D_OUT><<<(D_OUT * NPAD + TPB - 1) / TPB, TPB, 0, stream>>>(w2_rel,  w2relp);
  pad_weight<D_OUT><<<(D_OUT * NPAD + TPB - 1) / TPB, TPB, 0, stream>>>(w2_root, w2rootp);
  pad_bias<<<1, TPB, 0, stream>>>(b1_rel, b1p);
  pad_bias<<<1, TPB, 0, stream>>>(b2_rel, b2p);

  const int gemm_blocks = (N_NODES / 16 + (TPB / 32) - 1) / (TPB / 32);  // 391

  // ---- Layer 1: agg = scatter(x); h1 = agg@w1_rel + b1 + x@w1_root --------
  zero_f32<<<2048, TPB, 0, stream>>>(agg, N_NODES * D_IN);
  {
    long long threads = (long long)N_EDGES * (D_IN / 4);
    aggregate<D_IN><<<(int)((threads + TPB - 1) / TPB), TPB, 0, stream>>>(x, src, dst, agg);
  }
  gemm_dual_wmma<D_IN><<<gemm_blocks, TPB, 0, stream>>>(agg, x, w1relp, w1rootp, b1p, h1);

  // ---- Layer 2: agg = scatter(h1); h2 = agg@w2_rel + b2 + h1@w2_root ------
  zero_f32<<<2048, TPB, 0, stream>>>(agg, N_NODES * D_OUT);
  {
    long long threads = (long long)N_EDGES * (D_OUT / 4);
    aggregate<D_OUT><<<(int)((threads + TPB - 1) / TPB), TPB, 0, stream>>>(h1, src, dst, agg);
  }
  gemm_dual_wmma<D_OUT><<<gemm_blocks, TPB, 0, stream>>>(agg, h1, w2relp, w2rootp, b2p, h2);

  // ---- Layer 3 (only rows 0..9 matter; linearity of the mean) -------------
  zero_f32<<<1, TPB, 0, stream>>>(mean_h, 2 * D_OUT);   // mean_h + sum_agg
  mean_front10<<<1, TPB, 0, stream>>>(h2, mean_h);
  agg_front10<<<(N_EDGES + TPB - 1) / TPB, TPB, 0, stream>>>(h2, src, dst, sum_agg);
  final_gemv<<<1, TPB, 0, stream>>>(sum_agg, mean_h, w3_rel, w3_root, b3_rel, out);
}